// GNNPrototype_15668040696096
// MI455X (gfx1250) — compile-verified
//
#include <hip/hip_runtime.h>

// GCN 2-layer forward for MI455X (gfx1250, wave32).
//   N=100000 nodes, E=1.6M edges, F=H=128, C=8.
// Strategy:
//   out[d] = dinv[d] * sum_{s in N(d) U {d}} dinv[s]*h[s] + b
//   -> p = dinv[n]*(x@W)   (fp32 WMMA, row-scale fused in epilogue)
//   -> a[d] = p[d] + sum_edges p[s]   (L2-resident float atomics)
//   -> h1 = relu(dinv*a + b)          (fused finalize)
// fp32 precision preserved via V_WMMA_F32_16X16X4_F32 (GEMM is not the
// bottleneck; the L2-resident scatter-add is).

typedef float v2f __attribute__((ext_vector_type(2)));
typedef float v8f __attribute__((ext_vector_type(8)));

#define THREADS 256

// ---------------------------------------------------------------------------
// Degree / normalization
// ---------------------------------------------------------------------------
__global__ void k_deg_init(float* __restrict__ deg, int n) {
    int i = blockIdx.x * blockDim.x + threadIdx.x;
    if (i < n) deg[i] = 1.0f;  // self-loop
}

__global__ void k_deg_accum(const int* __restrict__ dst, float* __restrict__ deg, int E) {
    int i = blockIdx.x * blockDim.x + threadIdx.x;
    if (i >= E) return;
    __builtin_prefetch(dst + i + 8192, 0, 0);   // global_prefetch_b8 on edge stream
    atomicAdd(deg + dst[i], 1.0f);
}

__global__ void k_rsqrt_inplace(float* __restrict__ deg, int n) {
    int i = blockIdx.x * blockDim.x + threadIdx.x;
    if (i < n) deg[i] = rsqrtf(fmaxf(deg[i], 1.0f));
}

// ---------------------------------------------------------------------------
// Row-scaled GEMM via fp32 WMMA:  P[n][c] = dinv[n] * sum_k X[n][k]*W[k][c]
//   KDIM : inner dim (128)
//   NCOLT: number of 16-wide output column tiles
//   WCOLS: true #cols of W (cols >= WCOLS zero-padded in LDS, stores guarded)
// One wave computes a full 16-row x (NCOLT*16)-col block.
// ---------------------------------------------------------------------------
template<int KDIM, int NCOLT, int WCOLS>
__global__ __launch_bounds__(THREADS)
void k_gemm_rowscale(const float* __restrict__ X, const float* __restrict__ W,
                     const float* __restrict__ dinv, float* __restrict__ P, int nrows)
{
    __shared__ float sW[KDIM * NCOLT * 16];
    const int tid = threadIdx.x;
    // Stage W into LDS once per workgroup (zero-pad phantom columns).
    for (int i = tid; i < KDIM * NCOLT * 16; i += THREADS) {
        const int kk = i / (NCOLT * 16);
        const int c  = i % (NCOLT * 16);
        sW[i] = (c < WCOLS) ? W[kk * WCOLS + c] : 0.0f;
    }
    __syncthreads();

    const int gwave   = (blockIdx.x * THREADS + tid) >> 5;
    const int rowBase = gwave * 16;
    if (rowBase >= nrows) return;            // wave-uniform (N % 16 == 0 here)

    const int lane = tid & 31;
    const int lh   = lane >> 4;              // 0: lanes 0-15, 1: lanes 16-31
    const int ln   = lane & 15;

    v8f acc[NCOLT] = {};                     // 16x16 fp32 C/D tiles

    // A layout (16x4 f32): VGPR0 = K=k (lanes 0-15) / K=k+2 (lanes 16-31),
    //                      VGPR1 = K=k+1           / K=k+3.
    const float* xrow = X + (size_t)(rowBase + ln) * KDIM + 2 * lh;
    for (int kb = 0; kb < KDIM; kb += 4) {
        const v2f A = *reinterpret_cast<const v2f*>(xrow + kb);
        // B layout (4x16 f32) mirrors A: lane half selects K pair.
        const float* w0 = &sW[(kb + 2 * lh) * (NCOLT * 16) + ln];
        const float* w1 = w0 + NCOLT * 16;
        #pragma unroll
        for (int t = 0; t < NCOLT; ++t) {
            v2f B;
            B.x = w0[t * 16];
            B.y = w1[t * 16];
            acc[t] = __builtin_amdgcn_wmma_f32_16x16x4_f32(
                false, A, false, B, (short)0, acc[t], false, false);
        }
    }

    // D layout: VGPR r holds (M = r + 8*lh, N = ln). Fuse dinv row scale.
    const float* dvp = dinv + rowBase + 8 * lh;
    float dv[8];
    #pragma unroll
    for (int r = 0; r < 8; ++r) dv[r] = dvp[r];

    #pragma unroll
    for (int t = 0; t < NCOLT; ++t) {
        const int col = t * 16 + ln;
        if (col < WCOLS) {
            #pragma unroll
            for (int r = 0; r < 8; ++r)
                P[(size_t)(rowBase + 8 * lh + r) * WCOLS + col] = dv[r] * acc[t][r];
        }
    }
}

// ---------------------------------------------------------------------------
// a = p  (self-loop init), vectorized copy of n4 float4s
// ---------------------------------------------------------------------------
__global__ void k_copy4(const float4* __restrict__ src, float4* __restrict__ dst, int n4) {
    int i = blockIdx.x * blockDim.x + threadIdx.x;
    if (i < n4) dst[i] = src[i];
}

// ---------------------------------------------------------------------------
// Edge scatter, D=128: one wave per edge, float4 gather + 4 atomics per lane.
// p and a are L2-resident (51 MB each vs 192 MB L2).
// ---------------------------------------------------------------------------
__global__ void k_scatter128(const float* __restrict__ p,
                             const int* __restrict__ src, const int* __restrict__ dst,
                             float* __restrict__ a, int E)
{
    const int t    = blockIdx.x * blockDim.x + threadIdx.x;
    const int e    = t >> 5;
    const int lane = t & 31;
    if (e >= E) return;
    __builtin_prefetch(src + e + 2048, 0, 0);
    const int s = src[e];
    const int d = dst[e];
    const float4 v = *reinterpret_cast<const float4*>(p + (size_t)s * 128 + lane * 4);
    float* ap = a + (size_t)d * 128 + lane * 4;
    atomicAdd(ap + 0, v.x);
    atomicAdd(ap + 1, v.y);
    atomicAdd(ap + 2, v.z);
    atomicAdd(ap + 3, v.w);
}

// Edge scatter, D=8: one thread per edge (2 float4 loads, 8 atomics).
__global__ void k_scatter8(const float* __restrict__ p,
                           const int* __restrict__ src, const int* __restrict__ dst,
                           float* __restrict__ a, int E)
{
    const int e = blockIdx.x * blockDim.x + threadIdx.x;
    if (e >= E) return;
    const int s = src[e];
    const int d = dst[e];
    const float4 v0 = *reinterpret_cast<const float4*>(p + (size_t)s * 8);
    const float4 v1 = *reinterpret_cast<const float4*>(p + (size_t)s * 8 + 4);
    float* ap = a + (size_t)d * 8;
    atomicAdd(ap + 0, v0.x); atomicAdd(ap + 1, v0.y);
    atomicAdd(ap + 2, v0.z); atomicAdd(ap + 3, v0.w);
    atomicAdd(ap + 4, v1.x); atomicAdd(ap + 5, v1.y);
    atomicAdd(ap + 6, v1.z); atomicAdd(ap + 7, v1.w);
}

// ---------------------------------------------------------------------------
// out[n][c] = (relu?) ( dinv[n]*a[n][c] + bias[c] )
// ---------------------------------------------------------------------------
template<int D, bool RELU>
__global__ void k_finalize(const float* __restrict__ a, const float* __restrict__ dinv,
                           const float* __restrict__ bias, float* __restrict__ out, int n)
{
    const int i = blockIdx.x * blockDim.x + threadIdx.x;
    if (i >= n * D) return;
    const int node = i / D;
    const int c    = i % D;
    float v = dinv[node] * a[i] + bias[c];
    if (RELU) v = fmaxf(v, 0.0f);
    out[i] = v;
}

// ---------------------------------------------------------------------------
extern "C" void kernel_launch(void* const* d_in, const int* in_sizes, int n_in,
                              void* d_out, int out_size, void* d_ws, size_t ws_size,
                              hipStream_t stream)
{
    constexpr int F = 128, H = 128, C = 8;
    const int N = in_sizes[0] / F;        // 100000
    const int E = in_sizes[1] / 2;        // 1600000

    const float* x   = (const float*)d_in[0];
    const int*   src = (const int*)  d_in[1];
    const int*   dst = src + E;
    const float* W1  = (const float*)d_in[2];
    const float* b1  = (const float*)d_in[3];
    const float* W2  = (const float*)d_in[4];
    const float* b2  = (const float*)d_in[5];
    float*       out = (float*)d_out;

    // Workspace layout (floats): [deg/dinv: N][pbuf: N*128][abuf: N*128]
    // pbuf: p1 then reused for h1.  abuf: a1, then reused as {p2, a2}.
    float* ws   = (float*)d_ws;
    float* dinv = ws;                          // in-place deg -> rsqrt
    float* pbuf = ws + N;
    float* abuf = pbuf + (size_t)N * H;
    float* p2   = abuf;                        // a1 is dead by the time p2 is written
    float* a2   = abuf + (size_t)N * C;

    const int gN   = (N + THREADS - 1) / THREADS;
    const int gE   = (E + THREADS - 1) / THREADS;
    const int gGemm = ((N / 16) * 32 + THREADS - 1) / THREADS;   // one wave / 16 rows
    const int gNH  = ((size_t)N * H + THREADS - 1) / THREADS;
    const int gNH4 = ((size_t)N * H / 4 + THREADS - 1) / THREADS;
    const int gE32 = ((size_t)E * 32 + THREADS - 1) / THREADS;
    const int gNC  = ((size_t)N * C + THREADS - 1) / THREADS;
    const int gNC4 = ((size_t)N * C / 4 + THREADS - 1) / THREADS;

    // 1) symmetric normalization coefficients
    k_deg_init     <<<gN, THREADS, 0, stream>>>(dinv, N);
    k_deg_accum    <<<gE, THREADS, 0, stream>>>(dst, dinv, E);
    k_rsqrt_inplace<<<gN, THREADS, 0, stream>>>(dinv, N);

    // 2) layer 1: p1 = dinv * (x @ W1)
    k_gemm_rowscale<F, H / 16, H><<<gGemm, THREADS, 0, stream>>>(x, W1, dinv, pbuf, N);

    // 3) a1 = p1 (self loops) + scatter over edges
    k_copy4    <<<gNH4, THREADS, 0, stream>>>((const float4*)pbuf, (float4*)abuf, (int)((size_t)N * H / 4));
    k_scatter128<<<gE32, THREADS, 0, stream>>>(pbuf, src, dst, abuf, E);

    // 4) h1 = relu(dinv*a1 + b1)  -> overwrite pbuf
    k_finalize<H, true><<<gNH, THREADS, 0, stream>>>(abuf, dinv, b1, pbuf, N);

    // 5) layer 2: p2 = dinv * (h1 @ W2)   (W2 zero-padded to one 16-col tile)
    k_gemm_rowscale<H, 1, C><<<gGemm, THREADS, 0, stream>>>(pbuf, W2, dinv, p2, N);

    // 6) a2 = p2 + scatter
    k_copy4  <<<gNC4, THREADS, 0, stream>>>((const float4*)p2, (float4*)a2, (int)((size_t)N * C / 4));
    k_scatter8<<<gE, THREADS, 0, stream>>>(p2, src, dst, a2, E);

    // 7) out = dinv*a2 + b2
    k_finalize<C, false><<<gNC, THREADS, 0, stream>>>(a2, dinv, b2, out, N);
}